// JointRec_40492951666874
// MI455X (gfx1250) — compile-verified
//
#include <hip/hip_runtime.h>
#include <math.h>

typedef __attribute__((ext_vector_type(2))) float v2f;
typedef __attribute__((ext_vector_type(8))) float v8f;

#define B_TOTAL 8192
#define S_LEN   128
#define D_DIM   64
#define ROWS    16          // rows of B handled per block
#define ZS      130         // padded z row stride (conflict-free b64 frag loads)
#define GS      65          // padded g_stage row stride (conflict-free lane-per-s)

__device__ __forceinline__ float softplusf(float x) {
  // stable log(1+exp(x)), matches jax.nn.softplus
  return fmaxf(x, 0.0f) + log1pf(__expf(-fabsf(x)));
}

// Hardware tanh (CDNA5 V_TANH_F32, TRANS op co-executing with VALU).
__device__ __forceinline__ float fast_tanhf(float x) {
#if __has_builtin(__builtin_amdgcn_tanhf)
  return __builtin_amdgcn_tanhf(x);
#else
  float r;
  // v_nop covers the TRANS co-execution hazard (ISA 7.4: 1 indep op / V_NOP
  // required before the TRANS result is consumed).
  asm volatile("v_tanh_f32 %0, %1\n\tv_nop" : "=v"(r) : "v"(x));
  return r;
#endif
}

// ---------------- Kernel P: batch-independent stage precompute -------------
// stage_embed = relu(relu(feat@sW1)@sW2); g_stage = stage_embed@aW1[D:] + ab1
// closed[s] = 1 - exp(-beta*(S-s)); beta = softplus(intensity_decay)
__global__ void stage_precompute(
    const float* __restrict__ sW1, const float* __restrict__ sW2,
    const float* __restrict__ aW1, const float* __restrict__ ab1,
    const float* __restrict__ intensity_decay,
    float* __restrict__ ws_gstage,   // [S*D]
    float* __restrict__ ws_closed,   // [S]
    float* __restrict__ ws_beta)     // [1]
{
  int s = threadIdx.x;               // 0..127
  float fs = (float)s;
  float f0 = fs, f1 = fs * fs, f2 = log1pf(fs);

  float hid[32];
  #pragma unroll
  for (int j = 0; j < 32; ++j) {
    float h = f0 * sW1[0 * 32 + j] + f1 * sW1[1 * 32 + j] + f2 * sW1[2 * 32 + j];
    hid[j] = fmaxf(h, 0.0f);
  }
  float se[64];
  for (int k = 0; k < 64; ++k) {
    float acc = 0.f;
    #pragma unroll 8
    for (int j = 0; j < 32; ++j) acc = fmaf(hid[j], sW2[j * 64 + k], acc);
    se[k] = fmaxf(acc, 0.f);
  }
  for (int k = 0; k < 64; ++k) {
    float acc = ab1[k];
    #pragma unroll 8
    for (int kp = 0; kp < 64; ++kp) acc = fmaf(se[kp], aW1[(64 + kp) * 64 + k], acc);
    ws_gstage[s * 64 + k] = acc;
  }
  float beta = softplusf(intensity_decay[0]);
  ws_closed[s] = 1.0f - __expf(-beta * (float)(S_LEN - s));
  if (s == 0) ws_beta[0] = beta;
}

// ---------------- Kernel M: main per-batch kernel --------------------------
__global__ void __launch_bounds__(256)
joint_main(
    const int*   __restrict__ sub_x,
    const float* __restrict__ sub_pop,
    const float* __restrict__ mask,
    const float* __restrict__ user_emb,
    const float* __restrict__ item_emb,
    const float* __restrict__ iW1, const float* __restrict__ ib1,
    const float* __restrict__ iW2,
    const float* __restrict__ aW1, const float* __restrict__ aW2,
    const float* __restrict__ yW1, const float* __restrict__ yb1,
    const float* __restrict__ yW2,
    const float* __restrict__ user_conformity,
    const float* __restrict__ ws_gstage,
    const float* __restrict__ ws_closed,
    const float* __restrict__ ws_beta,
    float* __restrict__ out)
{
  __shared__ float zs[ROWS * ZS];      // z = [ue | ie], 16 x 128 (pad 130)
  __shared__ float gsl[S_LEN * GS];    // a_stage + ab1, 128 x 64 (pad 65)
  __shared__ float ai[ROWS * D_DIM];   // a_item tile, 16 x 64
  __shared__ float iW1s[D_DIM * 32];   // intensity W1, 64 x 32
  __shared__ float aW2s[D_DIM];
  __shared__ float closed_s[S_LEN];
  __shared__ float ylds[ROWS];
  __shared__ float int_lds[ROWS];

  const int t  = threadIdx.x;
  const int b0 = blockIdx.x * ROWS;

  // ---- Phase 0: cooperative gathers / staging (float4 where possible) ----
  for (int idx = t; idx < ROWS * 32; idx += 256) {  // 16 rows x 32 quads
    int row = idx >> 5, q = idx & 31;
    int col = q * 4;
    int b = b0 + row;
    float4 v;
    if (col < D_DIM)
      v = *(const float4*)&user_emb[sub_x[b * 3 + 0] * D_DIM + col];
    else
      v = *(const float4*)&item_emb[sub_x[b * 3 + 1] * D_DIM + (col - D_DIM)];
    zs[row * ZS + col + 0] = v.x;
    zs[row * ZS + col + 1] = v.y;
    zs[row * ZS + col + 2] = v.z;
    zs[row * ZS + col + 3] = v.w;
  }
  for (int idx = t; idx < (S_LEN * D_DIM) / 4; idx += 256) {
    float4 v = ((const float4*)ws_gstage)[idx];
    int s = idx >> 4, kq = (idx & 15) * 4;
    gsl[s * GS + kq + 0] = v.x;
    gsl[s * GS + kq + 1] = v.y;
    gsl[s * GS + kq + 2] = v.z;
    gsl[s * GS + kq + 3] = v.w;
  }
  for (int idx = t; idx < D_DIM * 32; idx += 256) iW1s[idx] = iW1[idx];
  if (t < D_DIM) aW2s[t] = aW2[t];
  if (t < S_LEN) closed_s[t] = ws_closed[t];
  __syncthreads();

  const int wave = t >> 5;
  const int lane = t & 31;
  const int lo = lane & 15, hi = lane >> 4;

  // ---- Phase 1: wave-specialized dense GEMMs (f32 WMMA 16x16x4) ----
  if (wave == 0) {
    // a_item = ie @ aW1[0:64]  (16x64 tile)
    for (int nt = 0; nt < 4; ++nt) {
      v8f acc = {0, 0, 0, 0, 0, 0, 0, 0};
      #pragma unroll 4
      for (int kb = 0; kb < 16; ++kb) {
        int ka = kb * 4 + 2 * hi;
        v2f a, bf;
        a.x  = zs[lo * ZS + D_DIM + ka];
        a.y  = zs[lo * ZS + D_DIM + ka + 1];
        bf.x = aW1[ka       * D_DIM + nt * 16 + lo];
        bf.y = aW1[(ka + 1) * D_DIM + nt * 16 + lo];
        acc = __builtin_amdgcn_wmma_f32_16x16x4_f32(
            false, a, false, bf, (short)0, acc, false, false);
      }
      #pragma unroll
      for (int r = 0; r < 8; ++r)
        ai[(r + 8 * hi) * D_DIM + nt * 16 + lo] = acc[r];
    }
  } else if (wave == 1) {
    // y = relu(z @ yW1 + yb1) @ yW2   (K = 128)
    float part[8] = {0, 0, 0, 0, 0, 0, 0, 0};
    for (int nt = 0; nt < 4; ++nt) {
      v8f acc = {0, 0, 0, 0, 0, 0, 0, 0};
      #pragma unroll 4
      for (int kb = 0; kb < 32; ++kb) {
        int ka = kb * 4 + 2 * hi;
        v2f a, bf;
        a.x  = zs[lo * ZS + ka];
        a.y  = zs[lo * ZS + ka + 1];
        bf.x = yW1[ka       * D_DIM + nt * 16 + lo];
        bf.y = yW1[(ka + 1) * D_DIM + nt * 16 + lo];
        acc = __builtin_amdgcn_wmma_f32_16x16x4_f32(
            false, a, false, bf, (short)0, acc, false, false);
      }
      int n = nt * 16 + lo;
      float w2 = yW2[n], bb = yb1[n];
      #pragma unroll
      for (int r = 0; r < 8; ++r)
        part[r] += fmaxf(acc[r] + bb, 0.f) * w2;
    }
    #pragma unroll
    for (int off = 8; off >= 1; off >>= 1)
      #pragma unroll
      for (int r = 0; r < 8; ++r)
        part[r] += __shfl_xor(part[r], off, 32);
    if (lo == 0)
      #pragma unroll
      for (int r = 0; r < 8; ++r) ylds[r + 8 * hi] = part[r];
  } else if (wave == 2) {
    // intensity = softplus(relu(ie@iW1 + ib1) @ iW2); lane pair per row
    int m = lo, jbase = hi * 16;
    float acc = 0.f;
    for (int j = 0; j < 16; ++j) {
      int jj = jbase + j;
      float h = ib1[jj];
      #pragma unroll 8
      for (int k = 0; k < D_DIM; ++k)
        h = fmaf(zs[m * ZS + D_DIM + k], iW1s[k * 32 + jj], h);
      acc = fmaf(fmaxf(h, 0.f), iW2[jj], acc);
    }
    acc += __shfl_xor(acc, 16, 32);
    if (hi == 0) int_lds[m] = softplusf(acc);
  }
  __syncthreads();

  // ---- Phase 2: s-loop (67M tanh across grid); wave -> 2 rows, lane -> s ----
  const float beta = ws_beta[0];

  // aW2 hoisted to registers (fully-unrolled constant indexing -> no scratch)
  float w2r[D_DIM];
  #pragma unroll
  for (int k = 0; k < D_DIM; ++k) w2r[k] = aW2s[k];

  for (int rb = 0; rb < 2; ++rb) {
    int row = wave * 2 + rb;
    int b = b0 + row;

    // a_item row hoisted to registers (uniform LDS broadcast, b128-friendly)
    float air[D_DIM];
    #pragma unroll
    for (int k = 0; k < D_DIM; ++k) air[k] = ai[row * D_DIM + k];

    float pop[4], mk[4], amp[4];
    float msum = 0.f, psum = 0.f, popsum = 0.f;
    #pragma unroll
    for (int c = 0; c < 4; ++c) {
      int s = c * 32 + lane;
      pop[c] = sub_pop[b * S_LEN + s];   // coalesced (lane == consecutive s)
      mk[c]  = mask[b * S_LEN + s];
      float dot = 0.f;
      #pragma unroll
      for (int k = 0; k < D_DIM; ++k) {
        float hv = fast_tanhf(air[k] + gsl[s * GS + k]);
        dot = fmaf(hv, w2r[k], dot);
      }
      amp[c] = softplusf(fast_tanhf(dot));
      msum   += mk[c];
      popsum += pop[c];
      psum = fmaf(pop[c] * amp[c], closed_s[s], psum);
    }
    #pragma unroll
    for (int off = 16; off >= 1; off >>= 1) msum += __shfl_xor(msum, off, 32);
    float last = msum - 1.0f;
    float dsum = 0.f;
    #pragma unroll
    for (int c = 0; c < 4; ++c) {
      int s = c * 32 + lane;
      float ek = beta * __expf(-beta * (last - (float)s));
      dsum = fmaf(pop[c] * mk[c] * amp[c], ek, dsum);
    }
    #pragma unroll
    for (int off = 16; off >= 1; off >>= 1) {
      dsum   += __shfl_xor(dsum, off, 32);
      psum   += __shfl_xor(psum, off, 32);
      popsum += __shfl_xor(popsum, off, 32);
    }
    if (lane == 0) {
      float inten     = int_lds[row];
      float density   = inten + dsum;
      float partition = inten * (float)S_LEN + psum;
      float lptv  = __logf(density / partition);
      float lputv = ylds[row] + lptv * user_conformity[0];
      float lpv   = __logf(popsum * (1.0f / 1000000.0f));
      out[b]               = lputv;
      out[B_TOTAL + b]     = lptv;
      out[2 * B_TOTAL + b] = lpv;
    }
  }
}

extern "C" void kernel_launch(void* const* d_in, const int* in_sizes, int n_in,
                              void* d_out, int out_size, void* d_ws, size_t ws_size,
                              hipStream_t stream) {
  (void)in_sizes; (void)n_in; (void)out_size; (void)ws_size;
  const int*   sub_x    = (const int*)d_in[0];
  const float* sub_pop  = (const float*)d_in[1];
  const float* mask     = (const float*)d_in[2];
  const float* user_emb = (const float*)d_in[3];
  const float* item_emb = (const float*)d_in[4];
  const float* sW1      = (const float*)d_in[5];
  const float* sW2      = (const float*)d_in[6];
  const float* iW1      = (const float*)d_in[7];
  const float* ib1      = (const float*)d_in[8];
  const float* iW2      = (const float*)d_in[9];
  const float* aW1      = (const float*)d_in[10];
  const float* ab1      = (const float*)d_in[11];
  const float* aW2      = (const float*)d_in[12];
  const float* yW1      = (const float*)d_in[13];
  const float* yb1      = (const float*)d_in[14];
  const float* yW2      = (const float*)d_in[15];
  const float* idec     = (const float*)d_in[16];
  const float* conf     = (const float*)d_in[17];

  float* ws        = (float*)d_ws;
  float* ws_gstage = ws;                         // S*D = 8192 floats
  float* ws_closed = ws + S_LEN * D_DIM;         // 128 floats
  float* ws_beta   = ws + S_LEN * D_DIM + S_LEN; // 1 float

  hipLaunchKernelGGL(stage_precompute, dim3(1), dim3(S_LEN), 0, stream,
                     sW1, sW2, aW1, ab1, idec, ws_gstage, ws_closed, ws_beta);
  hipLaunchKernelGGL(joint_main, dim3(B_TOTAL / ROWS), dim3(256), 0, stream,
                     sub_x, sub_pop, mask, user_emb, item_emb,
                     iW1, ib1, iW2, aW1, aW2, yW1, yb1, yW2, conf,
                     ws_gstage, ws_closed, ws_beta, (float*)d_out);
}